// SVDRegistrator_46145128628500
// MI455X (gfx1250) — compile-verified
//
#include <hip/hip_runtime.h>
#include <hip/hip_bf16.h>
#include <math.h>

typedef __attribute__((ext_vector_type(2))) float v2f;
typedef __attribute__((ext_vector_type(8))) float v8f;

constexpr int kB       = 32;
constexpr int kM       = 131072;
constexpr int kNB      = 32;            // partial records per batch (multiple of 4)
constexpr int kChunk   = kM / kNB;      // 4096 points per block
constexpr int kThreads = 256;           // 8 waves (wave32)

// ---------------------------------------------------------------------------
// Stage 1: streaming reduction. Each block accumulates the 4x4 homogeneous
// outer-product sum H = sum_n w_n * (a,1)(b,1)^T over its chunk of one batch,
// and writes a 16-float record to the workspace. Pure bandwidth kernel.
// ---------------------------------------------------------------------------
__global__ __launch_bounds__(kThreads)
void kabsch_reduce(const float* __restrict__ src,
                   const float* __restrict__ tgt,
                   const float* __restrict__ wts,
                   float* __restrict__ partials) {
  const int rec = blockIdx.x;
  const int b   = rec / kNB;
  const int blk = rec % kNB;
  const size_t p0 = (size_t)b * kM + (size_t)blk * kChunk;
  const float* __restrict__ A  = src + p0 * 3;
  const float* __restrict__ Bp = tgt + p0 * 3;
  const float* __restrict__ Wp = wts + p0;

  float s[16];
#pragma unroll
  for (int j = 0; j < 16; ++j) s[j] = 0.0f;

  for (int i = threadIdx.x; i < kChunk; i += kThreads) {
    const float ax = A[3 * i + 0], ay = A[3 * i + 1], az = A[3 * i + 2];
    const float bx = Bp[3 * i + 0], by = Bp[3 * i + 1], bz = Bp[3 * i + 2];
    const float w  = Wp[i];
    const float p[4] = { w * ax, w * ay, w * az, w };
    const float q[4] = { bx, by, bz, 1.0f };
#pragma unroll
    for (int ii = 0; ii < 4; ++ii)
#pragma unroll
      for (int jj = 0; jj < 4; ++jj)
        s[ii * 4 + jj] = fmaf(p[ii], q[jj], s[ii * 4 + jj]);
  }

  // wave32 butterfly reduction (all lanes active: kChunk % kThreads == 0)
#pragma unroll
  for (int j = 0; j < 16; ++j) {
#pragma unroll
    for (int off = 16; off > 0; off >>= 1)
      s[j] += __shfl_xor(s[j], off, 32);
  }

  __shared__ float sm[kThreads / 32][16];
  const int wave = threadIdx.x >> 5;
  const int lane = threadIdx.x & 31;
  if (lane == 0) {
#pragma unroll
    for (int j = 0; j < 16; ++j) sm[wave][j] = s[j];
  }
  __syncthreads();
  if (threadIdx.x < 16) {
    float acc = 0.0f;
#pragma unroll
    for (int wv = 0; wv < kThreads / 32; ++wv) acc += sm[wv][threadIdx.x];
    partials[(size_t)rec * 16 + threadIdx.x] = acc;
  }
}

// ---------------------------------------------------------------------------
// Stage 2: one wave per batch. Reduce kNB 16-float records with chained
// V_WMMA_F32_16X16X4_F32 (A = 16 sums x 4 records, B = ones(4x16)), then do
// the 3x3 SVD (Jacobi on cov^T cov) + det fix + translation on lane 0.
// ---------------------------------------------------------------------------
__global__ __launch_bounds__(32)
void kabsch_finalize(const float* __restrict__ partials,
                     float* __restrict__ out) {
  const int b    = blockIdx.x;
  const int lane = threadIdx.x;      // full wave, EXEC all ones for WMMA
  const int m    = lane & 15;
  const int half = lane >> 4;
  const float* __restrict__ base = partials + (size_t)b * kNB * 16;

  v8f acc = {0.f, 0.f, 0.f, 0.f, 0.f, 0.f, 0.f, 0.f};
  v2f ones;  ones.x = 1.0f;  ones.y = 1.0f;   // B[k][n] = 1 everywhere

#pragma unroll
  for (int c = 0; c < kNB; c += 4) {
    // ISA 16x4 f32 A layout: lane m holds A[m][0],A[m][1]; lane m+16 holds
    // A[m][2],A[m][3]. Set A[m][k] = record (c+k), sum index m.
    v2f a;
    a.x = base[(size_t)(c + 2 * half + 0) * 16 + m];
    a.y = base[(size_t)(c + 2 * half + 1) * 16 + m];
    // D[m][n] += sum_k A[m][k] * 1  -> every column of row m = running total
    acc = __builtin_amdgcn_wmma_f32_16x16x4_f32(false, a, false, ones,
                                                (short)0, acc, false, false);
  }

  // D layout: VGPR v, lanes 0-15 -> row v; lanes 16-31 -> row v+8.
  float T[16];
#pragma unroll
  for (int j = 0; j < 8; ++j) {
    T[j]     = __shfl(acc[j], 0, 32);
    T[8 + j] = __shfl(acc[j], 16, 32);
  }

  if (lane != 0) return;

  // ----- covariance from homogeneous sums (matches reference normalization)
  const float Wn  = T[15] + 1e-5f;          // sum(w) + EPS
  const float inv = 1.0f / Wn;
  const float cA[3] = { T[3] * inv, T[7] * inv, T[11] * inv };
  const float cB[3] = { T[12] * inv, T[13] * inv, T[14] * inv };
  const float fac = 2.0f - T[15] * inv;     // 2 - sum(w_norm)
  float C[3][3];
#pragma unroll
  for (int i = 0; i < 3; ++i)
#pragma unroll
    for (int j = 0; j < 3; ++j)
      C[i][j] = T[i * 4 + j] * inv - cA[i] * cB[j] * fac;

  // ----- N = C^T C, Jacobi eigendecomposition -> V, eigenvalues
  float N[3][3];
  for (int i = 0; i < 3; ++i)
    for (int j = 0; j < 3; ++j)
      N[i][j] = C[0][i] * C[0][j] + C[1][i] * C[1][j] + C[2][i] * C[2][j];

  float V[3][3] = { {1.f, 0.f, 0.f}, {0.f, 1.f, 0.f}, {0.f, 0.f, 1.f} };
  for (int sweep = 0; sweep < 8; ++sweep) {
    for (int pair = 0; pair < 3; ++pair) {
      const int p = (pair == 2) ? 1 : 0;
      const int q = (pair == 0) ? 1 : 2;
      const float apq = N[p][q];
      if (fabsf(apq) < 1e-30f) continue;
      const float tau = (N[q][q] - N[p][p]) / (2.0f * apq);
      const float t   = ((tau >= 0.0f) ? 1.0f : -1.0f) /
                        (fabsf(tau) + sqrtf(1.0f + tau * tau));
      const float cc = 1.0f / sqrtf(1.0f + t * t);
      const float ss = t * cc;
      for (int k = 0; k < 3; ++k) {           // N = N * J
        const float nkp = N[k][p], nkq = N[k][q];
        N[k][p] = cc * nkp - ss * nkq;
        N[k][q] = ss * nkp + cc * nkq;
      }
      for (int k = 0; k < 3; ++k) {           // N = J^T * N
        const float npk = N[p][k], nqk = N[q][k];
        N[p][k] = cc * npk - ss * nqk;
        N[q][k] = ss * npk + cc * nqk;
      }
      for (int k = 0; k < 3; ++k) {           // V = V * J
        const float vkp = V[k][p], vkq = V[k][q];
        V[k][p] = cc * vkp - ss * vkq;
        V[k][q] = ss * vkp + cc * vkq;
      }
    }
  }

  // sort eigenpairs descending (columns of V)
  float lam[3] = { N[0][0], N[1][1], N[2][2] };
  for (int pass = 0; pass < 2; ++pass) {
    for (int c0 = 0; c0 < 2; ++c0) {
      if (lam[c0] < lam[c0 + 1]) {
        const float tl = lam[c0]; lam[c0] = lam[c0 + 1]; lam[c0 + 1] = tl;
        for (int k = 0; k < 3; ++k) {
          const float tv = V[k][c0]; V[k][c0] = V[k][c0 + 1]; V[k][c0 + 1] = tv;
        }
      }
    }
  }

  // ----- U columns: u0 = normalize(C v0); u1 orthonormalized; u2 = +-cross
  float u0[3], u1[3], u2[3], t2[3];
  for (int i = 0; i < 3; ++i) {
    u0[i] = C[i][0] * V[0][0] + C[i][1] * V[1][0] + C[i][2] * V[2][0];
    u1[i] = C[i][0] * V[0][1] + C[i][1] * V[1][1] + C[i][2] * V[2][1];
    t2[i] = C[i][0] * V[0][2] + C[i][1] * V[1][2] + C[i][2] * V[2][2];
  }
  float n0 = sqrtf(u0[0] * u0[0] + u0[1] * u0[1] + u0[2] * u0[2]);
  if (n0 > 1e-30f) { const float r = 1.0f / n0; u0[0] *= r; u0[1] *= r; u0[2] *= r; }
  else { u0[0] = 1.0f; u0[1] = 0.0f; u0[2] = 0.0f; }
  const float d01 = u1[0] * u0[0] + u1[1] * u0[1] + u1[2] * u0[2];
  for (int i = 0; i < 3; ++i) u1[i] -= d01 * u0[i];
  float n1 = sqrtf(u1[0] * u1[0] + u1[1] * u1[1] + u1[2] * u1[2]);
  if (n1 > 1e-30f) { const float r = 1.0f / n1; u1[0] *= r; u1[1] *= r; u1[2] *= r; }
  else {  // any unit vector orthogonal to u0
    const float ax = fabsf(u0[0]);
    if (ax < 0.9f) { u1[0] = 0.f; u1[1] = -u0[2]; u1[2] = u0[1]; }
    else           { u1[0] = -u0[2]; u1[1] = 0.f; u1[2] = u0[0]; }
    const float r = 1.0f / sqrtf(u1[0]*u1[0] + u1[1]*u1[1] + u1[2]*u1[2]);
    u1[0] *= r; u1[1] *= r; u1[2] *= r;
  }
  u2[0] = u0[1] * u1[2] - u0[2] * u1[1];
  u2[1] = u0[2] * u1[0] - u0[0] * u1[2];
  u2[2] = u0[0] * u1[1] - u0[1] * u1[0];
  if (u2[0] * t2[0] + u2[1] * t2[1] + u2[2] * t2[2] < 0.0f) {
    u2[0] = -u2[0]; u2[1] = -u2[1]; u2[2] = -u2[2];   // match C v2 = s2*u2 sign
  }

  // ----- rot = V U^T, det fix (negate V column 2), translation
  float R[3][3];
  for (int i = 0; i < 3; ++i)
    for (int j = 0; j < 3; ++j)
      R[i][j] = V[i][0] * u0[j] + V[i][1] * u1[j] + V[i][2] * u2[j];
  const float det =
      R[0][0] * (R[1][1] * R[2][2] - R[1][2] * R[2][1]) -
      R[0][1] * (R[1][0] * R[2][2] - R[1][2] * R[2][0]) +
      R[0][2] * (R[1][0] * R[2][1] - R[1][1] * R[2][0]);
  if (det < 0.0f) {
    for (int i = 0; i < 3; ++i)
      for (int j = 0; j < 3; ++j)
        R[i][j] = V[i][0] * u0[j] + V[i][1] * u1[j] - V[i][2] * u2[j];
  }

  float* o = out + (size_t)b * 12;
  for (int i = 0; i < 3; ++i) {
    const float tr = -(R[i][0] * cA[0] + R[i][1] * cA[1] + R[i][2] * cA[2]) + cB[i];
    o[i * 4 + 0] = R[i][0];
    o[i * 4 + 1] = R[i][1];
    o[i * 4 + 2] = R[i][2];
    o[i * 4 + 3] = tr;
  }
}

// ---------------------------------------------------------------------------
extern "C" void kernel_launch(void* const* d_in, const int* in_sizes, int n_in,
                              void* d_out, int out_size, void* d_ws, size_t ws_size,
                              hipStream_t stream) {
  const float* src = (const float*)d_in[0];   // [B, M, 3] f32
  const float* tgt = (const float*)d_in[1];   // [B, M, 3] f32
  const float* wts = (const float*)d_in[2];   // [B, M]    f32
  float* out = (float*)d_out;                 // [B, 3, 4] f32
  float* ws  = (float*)d_ws;                  // kB*kNB*16 floats = 64 KB

  kabsch_reduce<<<dim3(kB * kNB), dim3(kThreads), 0, stream>>>(src, tgt, wts, ws);
  kabsch_finalize<<<dim3(kB), dim3(32), 0, stream>>>(ws, out);
}